// GATAggregator_23510650978752
// MI455X (gfx1250) — compile-verified
//
#include <hip/hip_runtime.h>
#include <math.h>

#define DIM 128
#define WAVES 8
#define CAP 512

typedef float v2f __attribute__((ext_vector_type(2)));
typedef float v8f __attribute__((ext_vector_type(8)));

__device__ __forceinline__ float wave_sum(float x) {
#pragma unroll
  for (int o = 16; o > 0; o >>= 1) x += __shfl_xor(x, o, 32);
  return x;
}
__device__ __forceinline__ float wave_max(float x) {
#pragma unroll
  for (int o = 16; o > 0; o >>= 1) x = fmaxf(x, __shfl_xor(x, o, 32));
  return x;
}

// ---------------------------------------------------------------------------
// Kernel 1: v = W^T a  (the only true matrix op after algebraic folding).
// One wave, fp32 WMMA 16x16x4: D_j += A * B,  A row0 = a[k..k+3], B = W tile.
// A layout (32b 16x4): VGPR0 lanes0-15 -> (M=lane,K=0), lanes16-31 -> (M,K=2);
//                      VGPR1 -> K=1 / K=3.   B mirrors with N striped on lanes.
// Result row M=0 lives in D VGPR0, lanes 0..15.
// ---------------------------------------------------------------------------
#if __has_builtin(__builtin_amdgcn_wmma_f32_16x16x4_f32)
#define USE_WMMA_FOLD 1
__global__ void fold_attn_vec_wmma(const float* __restrict__ W,
                                   const float* __restrict__ a,
                                   float* __restrict__ v) {
  const int lane  = threadIdx.x;
  const int half  = lane >> 4;   // 0 -> K offset 0, 1 -> K offset +2
  const int lan16 = lane & 15;

  v8f acc[8];
#pragma unroll
  for (int j = 0; j < 8; ++j) acc[j] = (v8f)(0.0f);

  for (int kc = 0; kc < 32; ++kc) {
    const int k0 = kc * 4 + 2 * half;
    v2f A; A.x = 0.0f; A.y = 0.0f;
    if (lan16 == 0) { A.x = a[k0]; A.y = a[k0 + 1]; }
#pragma unroll
    for (int j = 0; j < 8; ++j) {
      v2f B;
      B.x = W[(size_t)k0 * DIM + j * 16 + lan16];
      B.y = W[(size_t)(k0 + 1) * DIM + j * 16 + lan16];
      acc[j] = __builtin_amdgcn_wmma_f32_16x16x4_f32(
          false, A, false, B, (short)0, acc[j], false, false);
    }
  }
  if (lane < 16) {
#pragma unroll
    for (int j = 0; j < 8; ++j) v[j * 16 + lane] = acc[j][0];
  }
}
#else
#define USE_WMMA_FOLD 0
#endif

__global__ void fold_attn_vec_scalar(const float* __restrict__ W,
                                     const float* __restrict__ a,
                                     float* __restrict__ v) {
  const int j = threadIdx.x;
  float acc = 0.0f;
  for (int d = 0; d < DIM; ++d) acc += a[d] * W[(size_t)d * DIM + j];
  v[j] = acc;
}

// ---------------------------------------------------------------------------
// Kernel 2: one wave per segment. Sorted segment_ids -> contiguous ranges.
// ---------------------------------------------------------------------------
__device__ __forceinline__ int lower_bound_i32(const int* __restrict__ a,
                                               int n, int key) {
  int lo = 0, hi = n;
  while (lo < hi) {
    int mid = (lo + hi) >> 1;
    if (a[mid] < key) lo = mid + 1; else hi = mid;
  }
  return lo;
}

__global__ void gat_segment_kernel(const float* __restrict__ messages,
                                   const float* __restrict__ tstamps,
                                   const int* __restrict__ seg,
                                   const float* __restrict__ v,
                                   float* __restrict__ out_agg,
                                   float* __restrict__ out_ts,
                                   int E, int N) {
  __shared__ float sc_buf[WAVES][CAP];
  const int lane = threadIdx.x & 31;
  const int wid  = threadIdx.x >> 5;
  const int s    = blockIdx.x * WAVES + wid;
  if (s >= N) return;  // uniform per-wave exit; no block barriers used
  float* sc = sc_buf[wid];

  const int start = lower_bound_i32(seg, E, s);
  const int end   = lower_bound_i32(seg, E, s + 1);
  const int len   = end - start;

  float4* aggrow = reinterpret_cast<float4*>(out_agg + (size_t)s * DIM);

  if (len == 0) {
    aggrow[lane] = make_float4(0.f, 0.f, 0.f, 0.f);
    if (lane == 0) out_ts[s] = -INFINITY;
    return;
  }

  // segment max of timestamps
  float tmax = -INFINITY;
  for (int i = lane; i < len; i += 32) tmax = fmaxf(tmax, tstamps[start + i]);
  tmax = wave_max(tmax);

  // per-lane float4 slice of folded vector v
  const float4 v4 = reinterpret_cast<const float4*>(v)[lane];

  // ---- pass 1: scores + running max (scores cached in LDS up to CAP) ----
  float m = -INFINITY;
  for (int e = 0; e < len; ++e) {
    const float4 m4 =
        reinterpret_cast<const float4*>(messages + (size_t)(start + e) * DIM)[lane];
    float p = m4.x * v4.x + m4.y * v4.y + m4.z * v4.z + m4.w * v4.w;
    float scv = wave_sum(p);                       // all lanes get the dot
    scv = scv > 0.0f ? scv : 0.2f * scv;           // leaky_relu(0.2)
    if (e < CAP && lane == 0) sc[e] = scv;
    m = fmaxf(m, scv);
  }
  __builtin_amdgcn_wave_barrier();  // keep LDS stores before the sweep

  // ---- sum of exp: cached part lane-parallel, converts sc[] -> exp in place
  const int clen = len < CAP ? len : CAP;
  float spart = 0.0f;
  for (int i = lane; i < clen; i += 32) {
    float ev = __expf(sc[i] - m);
    sc[i] = ev;
    spart += ev;
  }
  float denom = wave_sum(spart);
  // rare tail (len > CAP): cooperative recompute, replicated accumulate
  for (int e = CAP; e < len; ++e) {
    const float4 m4 =
        reinterpret_cast<const float4*>(messages + (size_t)(start + e) * DIM)[lane];
    float p = m4.x * v4.x + m4.y * v4.y + m4.z * v4.z + m4.w * v4.w;
    float scv = wave_sum(p);
    scv = scv > 0.0f ? scv : 0.2f * scv;
    denom += __expf(scv - m);
  }
  const float inv = 1.0f / denom;
  __builtin_amdgcn_wave_barrier();

  // ---- pass 2: weighted sum of RAW messages (rows hot in WGP$/L2) ----
  float4 acc = make_float4(0.f, 0.f, 0.f, 0.f);
  for (int e = 0; e < len; ++e) {
    const float4 m4 =
        reinterpret_cast<const float4*>(messages + (size_t)(start + e) * DIM)[lane];
    float ev;
    if (e < CAP) {
      ev = sc[e];  // broadcast LDS read
    } else {
      float p = m4.x * v4.x + m4.y * v4.y + m4.z * v4.z + m4.w * v4.w;
      float scv = wave_sum(p);
      scv = scv > 0.0f ? scv : 0.2f * scv;
      ev = __expf(scv - m);
    }
    const float wgt = ev * inv;
    acc.x += m4.x * wgt;
    acc.y += m4.y * wgt;
    acc.z += m4.z * wgt;
    acc.w += m4.w * wgt;
  }
  aggrow[lane] = acc;
  if (lane == 0) out_ts[s] = tmax;
}

// ---------------------------------------------------------------------------
extern "C" void kernel_launch(void* const* d_in, const int* in_sizes, int n_in,
                              void* d_out, int out_size, void* d_ws, size_t ws_size,
                              hipStream_t stream) {
  const float* messages   = (const float*)d_in[0];
  const float* timestamps = (const float*)d_in[1];
  const int*   seg        = (const int*)d_in[2];
  // d_in[3] = num_segments scalar (device); N derived from out_size instead
  const float* W          = (const float*)d_in[4];
  const float* attn_vec   = (const float*)d_in[5];

  const int E = in_sizes[1];                 // timestamps length = #edges
  const int N = out_size / (DIM + 1);        // out = [N*128 agg] ++ [N ts]

  float* out_agg = (float*)d_out;
  float* out_ts  = (float*)d_out + (size_t)N * DIM;
  float* v       = (float*)d_ws;             // 128 floats of scratch

#if USE_WMMA_FOLD
  hipLaunchKernelGGL(fold_attn_vec_wmma, dim3(1), dim3(32), 0, stream,
                     W, attn_vec, v);
#else
  hipLaunchKernelGGL(fold_attn_vec_scalar, dim3(1), dim3(DIM), 0, stream,
                     W, attn_vec, v);
#endif

  const int blocks = (N + WAVES - 1) / WAVES;
  hipLaunchKernelGGL(gat_segment_kernel, dim3(blocks), dim3(WAVES * 32), 0,
                     stream, messages, timestamps, seg, v, out_agg, out_ts, E, N);
}